// RicciTensorNetwork_51376398795073
// MI455X (gfx1250) — compile-verified
//
#include <hip/hip_runtime.h>
#include <hip/hip_bf16.h>

typedef float v2f __attribute__((ext_vector_type(2)));
typedef float v8f __attribute__((ext_vector_type(8)));

// ---------------------------------------------------------------------------
// Stage A (single block): metric = L2(relu(L1(points))), plus per-b bases for
// the christoffel and ricci MLPs (points @ first-32-rows of cw1 / rw1 + bias).
// ---------------------------------------------------------------------------
__global__ __launch_bounds__(1024) void ricci_stage_a(
    const float* __restrict__ points,
    const float* __restrict__ mw1, const float* __restrict__ mb1,
    const float* __restrict__ mw2, const float* __restrict__ mb2,
    const float* __restrict__ cw1, const float* __restrict__ cb1,
    const float* __restrict__ rw1, const float* __restrict__ rb1,
    float* __restrict__ metric,   // [32*1024]
    float* __restrict__ cbase,    // [32*128]
    float* __restrict__ rbase)    // [32*256]
{
    __shared__ float h1[32 * 128];
    __shared__ float pts[32 * 32];
    const int tid = threadIdx.x;

    pts[tid] = points[tid];
    __syncthreads();

    // h1 = relu(points@mw1+mb1) and cbase = points@cw1[0:32]+cb1  (4096 each)
    for (int e = tid; e < 32 * 128; e += 1024) {
        const int b = e >> 7, h = e & 127;
        const float* pb = &pts[b * 32];
        float am = mb1[h];
        float ac = cb1[h];
        #pragma unroll 8
        for (int f = 0; f < 32; ++f) {
            const float p = pb[f];
            am = fmaf(p, mw1[f * 128 + h], am);
            ac = fmaf(p, cw1[f * 128 + h], ac);
        }
        h1[e]    = fmaxf(am, 0.0f);
        cbase[e] = ac;
    }
    // rbase = points@rw1[0:32] + rb1   (8192 elems)
    for (int e = tid; e < 32 * 256; e += 1024) {
        const int b = e >> 8, h = e & 255;
        const float* pb = &pts[b * 32];
        float ar = rb1[h];
        #pragma unroll 8
        for (int f = 0; f < 32; ++f)
            ar = fmaf(pb[f], rw1[f * 256 + h], ar);
        rbase[e] = ar;
    }
    __syncthreads();

    // metric = h1 @ mw2 + mb2   (32768 elems, 128-deep dot each)
    for (int e = tid; e < 32 * 1024; e += 1024) {
        const int b = e >> 10, m = e & 1023;
        const float* hb = &h1[b * 128];
        float acc = mb2[m];
        #pragma unroll 8
        for (int h = 0; h < 128; ++h)
            acc = fmaf(hb[h], mw2[h * 1024 + m], acc);
        metric[e] = acc;
    }
}

// ---------------------------------------------------------------------------
// Stage B: christoffel[b,i,j,k].  One wave per (b,i,j), lane = k.
//
// Factored layer 1:  t[h] = u[h] + gjk*w33'[h] + gki*w34'[h]
//   where u[h] = 2log2e*(cbase_b[h] + gij*w32[h])  (per-wave, staged in LDS)
//   and   w33'/w34' are pre-scaled by 2log2e      (per-block, staged in LDS)
// so exp(2t) = exp2(t') comes straight off two FMAs -> v_exp_f32.
//
// tanh elimination:  sum_h tanh*w2 = sum(w2) - 2*sum(rcp(e+1)*w2)
// Inner loop: 2 FMA + add + v_exp_f32 + v_rcp_f32 + FMA, 4 LDS reads.
// ---------------------------------------------------------------------------
__global__ __launch_bounds__(256) void ricci_stage_b(
    const float* __restrict__ metric, const float* __restrict__ cbase,
    const float* __restrict__ cw1, const float* __restrict__ cw2,
    const float* __restrict__ cb2p,
    float* __restrict__ gamma)     // [32^4]
{
    __shared__ float s_base[128], s_w32[128], s_w33[128], s_w34[128], s_w2[128];
    __shared__ float s_u[8][128];
    __shared__ float s_w2sum;
    const float C = 2.8853900817779268f;   // 2 * log2(e)

    const int tid  = threadIdx.x;
    const int wid0 = blockIdx.x * 8;       // 8 waves per block; b uniform per block
    const int b    = wid0 >> 10;

    if (tid < 128) {
        s_base[tid] = cbase[b * 128 + tid];
        s_w32[tid]  = cw1[32 * 128 + tid];
        s_w33[tid]  = C * cw1[33 * 128 + tid];
        s_w34[tid]  = C * cw1[34 * 128 + tid];
        s_w2[tid]   = cw2[tid];
    }
    __syncthreads();

    if (tid == 0) {
        float s = 0.0f;
        #pragma unroll 8
        for (int h = 0; h < 128; ++h) s += s_w2[h];
        s_w2sum = s;
    }

    const int wv   = tid >> 5;
    const int lane = tid & 31;             // = k
    const int wid  = wid0 + wv;
    const int i    = (wid >> 5) & 31;
    const int j    = wid & 31;

    const float* mb_ = metric + b * 1024;
    const float gij = mb_[i * 32 + j];
    const float gjk = mb_[j * 32 + lane];
    const float gki = mb_[lane * 32 + i];

    // per-wave u[] slice: fold wave-uniform gij term + exp2 scale
    #pragma unroll
    for (int q = 0; q < 4; ++q) {
        const int h = lane * 4 + q;
        s_u[wv][h] = C * fmaf(gij, s_w32[h], s_base[h]);
    }
    __syncthreads();   // covers s_u and s_w2sum

    float acc = 0.0f;  // sum of rcp(e+1) * w2
    #pragma unroll 4
    for (int h = 0; h < 128; ++h) {
        float t = fmaf(gki, s_w34[h], fmaf(gjk, s_w33[h], s_u[wv][h]));
        const float e = __builtin_amdgcn_exp2f(t);
        const float r = __builtin_amdgcn_rcpf(e + 1.0f);
        acc = fmaf(r, s_w2[h], acc);
    }
    // Gamma = sum(w2) - 2*acc + cb2
    gamma[wid * 32 + lane] = fmaf(-2.0f, acc, s_w2sum + cb2p[0]);
}

// ---------------------------------------------------------------------------
// Stage C: ricci[b,i,j].  Gamma-GEMM [32768x32]@[32x256] on the matrix core:
// each 512-thread block handles 16 samples (fixed b,i; j in [j0,j0+16)) x 256
// hidden; wave w owns N-tile [16w,16w+16), 8x v_wmma_f32_16x16x4_f32 over K=32.
// Fused: + rbase_b[n] + g_ij*rw1[32][n], ReLU, dot with rw2, reduce.
// ---------------------------------------------------------------------------
__global__ __launch_bounds__(512) void ricci_stage_c(
    const float* __restrict__ metric, const float* __restrict__ rbase,
    const float* __restrict__ gamma,
    const float* __restrict__ rw1, const float* __restrict__ rw2,
    const float* __restrict__ rb2p,
    float* __restrict__ out)       // [32*32*32]
{
    __shared__ float sums[16][17];
    const int tid  = threadIdx.x;
    const int lane = tid & 31;
    const int w    = tid >> 5;             // N-tile index (0..15)
    const int blk  = blockIdx.x;
    const int b    = blk >> 6;
    const int rem  = blk & 63;
    const int i    = rem >> 1;
    const int j0   = (rem & 1) << 4;

    const int n  = w * 16 + (lane & 15);   // hidden column for this lane
    const int hi = lane >> 4;              // 0: lanes 0-15, 1: lanes 16-31
    const int m  = lane & 15;              // A-fragment row (sample within tile)

    // A rows: gamma[b][i][j0+m][0..31]
    const float* Grow = gamma + (((b * 32 + i) * 32) + j0 + m) * 32;

    v8f acc = {};
    #pragma unroll
    for (int ks = 0; ks < 8; ++ks) {
        const int kb = ks * 4 + hi * 2;    // K split: low half-wave K{0,1}, high K{2,3}
        v2f a, bm;
        a.x  = Grow[kb];
        a.y  = Grow[kb + 1];
        bm.x = rw1[(33 + kb) * 256 + n];
        bm.y = rw1[(34 + kb) * 256 + n];
        acc = __builtin_amdgcn_wmma_f32_16x16x4_f32(
            /*neg_a=*/false, a, /*neg_b=*/false, bm,
            /*c_mod=*/(short)0, acc, /*reuse_a=*/false, /*reuse_b=*/false);
    }

    const float rb_n  = rbase[b * 256 + n];
    const float w32_n = rw1[32 * 256 + n];
    const float rw2_n = rw2[n];
    const float* mrowi = metric + (b * 32 + i) * 32;  // metric[b][i][*]

    float p[8];
    #pragma unroll
    for (int r = 0; r < 8; ++r) {
        const int row = r + hi * 8;                    // C/D layout: M = r (+8 hi half)
        const float gij = mrowi[j0 + row];
        float h = acc[r] + rb_n + gij * w32_n;
        h = fmaxf(h, 0.0f);
        p[r] = h * rw2_n;
    }
    // reduce over the 16 hidden columns held by this half-wave
    #pragma unroll
    for (int mask = 1; mask <= 8; mask <<= 1) {
        #pragma unroll
        for (int r = 0; r < 8; ++r) p[r] += __shfl_xor(p[r], mask, 32);
    }
    if ((lane & 15) == 0) {
        #pragma unroll
        for (int r = 0; r < 8; ++r) sums[w][r + hi * 8] = p[r];
    }
    __syncthreads();
    // cross-wave reduction: 16 waves x 16 rows
    if (tid < 16) {
        float tot = rb2p[0];
        #pragma unroll
        for (int ww = 0; ww < 16; ++ww) tot += sums[ww][tid];
        out[(b * 32 + i) * 32 + j0 + tid] = tot;
    }
}

// ---------------------------------------------------------------------------
extern "C" void kernel_launch(void* const* d_in, const int* in_sizes, int n_in,
                              void* d_out, int out_size, void* d_ws, size_t ws_size,
                              hipStream_t stream) {
    const float* points = (const float*)d_in[0];
    const float* mw1 = (const float*)d_in[1];
    const float* mb1 = (const float*)d_in[2];
    const float* mw2 = (const float*)d_in[3];
    const float* mb2 = (const float*)d_in[4];
    const float* cw1 = (const float*)d_in[5];
    const float* cb1 = (const float*)d_in[6];
    const float* cw2 = (const float*)d_in[7];
    const float* cb2 = (const float*)d_in[8];
    const float* rw1 = (const float*)d_in[9];
    const float* rb1 = (const float*)d_in[10];
    const float* rw2 = (const float*)d_in[11];
    const float* rb2 = (const float*)d_in[12];

    float* ws     = (float*)d_ws;
    float* metric = ws;                // 32*1024   = 32768
    float* cbase  = ws + 32768;        // 32*128    = 4096
    float* rbase  = ws + 36864;        // 32*256    = 8192
    float* gamma  = ws + 45056;        // 32^4      = 1048576  (~4.4 MB total)

    ricci_stage_a<<<1, 1024, 0, stream>>>(points, mw1, mb1, mw2, mb2,
                                          cw1, cb1, rw1, rb1,
                                          metric, cbase, rbase);
    ricci_stage_b<<<4096, 256, 0, stream>>>(metric, cbase, cw1, cw2, cb2, gamma);
    ricci_stage_c<<<2048, 512, 0, stream>>>(metric, rbase, gamma, rw1, rw2, rb2,
                                            (float*)d_out);
}